// MeanMessageAggregator_45681272160567
// MI455X (gfx1250) — compile-verified
//
#include <hip/hip_runtime.h>

// Segment-mean aggregation for MI455X (gfx1250, wave32).
//
//   sums : d_out  (N x 256 f32) -- accumulated in place, divided at the end
//   cnt  : d_ws   (N f32)
//
// Roofline: 512 MB NT-streamed reads + 128M no-return f32 atomics (resolve in
// the 192 MB L2, where the 102.8 MB accumulator+counts stay resident) +
// 102 MB NT-streamed output. ~26 us HBM floor; WMMA inapplicable (no dense
// contraction).

typedef float v4f __attribute__((ext_vector_type(4)));

#define TPB 256  // 8 wave32 waves per block; each wave = half a message row

// Defined FIRST so the disasm snippet shows the hot loop:
// expect global_load_b128 ... th:TH_LOAD_NT and 5x global_atomic_add_f32
// (no-return form) with scalar (saddr) base addressing.
__global__ void __launch_bounds__(TPB)
mean_agg_scatter(const float* __restrict__ M,
                 const int* __restrict__ nodes,
                 float* __restrict__ sums,
                 float* __restrict__ cnt,
                 unsigned total4) {
  const unsigned i = blockIdx.x * TPB + threadIdx.x;  // < 32M, 32-bit math
  if (i >= total4) return;

  const unsigned msg = i >> 6;   // 64 float4 per 256-wide row; wave-uniform
  const unsigned c4  = i & 63;

  // msg is uniform across each wave32 (block base is a multiple of 64,
  // each wave spans 32 consecutive float4 inside one row) -> scalarize the
  // node id so atomic addresses become SGPR-base + small per-lane offset.
  const int node = __builtin_amdgcn_readfirstlane(nodes[msg]);

  // Stream M past L2 (non-temporal): read-once data must not evict the
  // L2-resident accumulator.
  const v4f m = __builtin_nontemporal_load((const v4f*)M + i);

  float* dst = sums + (unsigned)node * 256u + (c4 << 2);
  // Native no-return hardware f32 atomics (STOREcnt-tracked), RMW in L2.
  unsafeAtomicAdd(dst + 0, m.x);
  unsafeAtomicAdd(dst + 1, m.y);
  unsafeAtomicAdd(dst + 2, m.z);
  unsafeAtomicAdd(dst + 3, m.w);

  if (c4 == 0) {
    unsafeAtomicAdd(cnt + (unsigned)node, 1.0f);
  }
}

__global__ void __launch_bounds__(TPB)
mean_agg_zero(float* __restrict__ sums, float* __restrict__ cnt,
              int out_n4, int n_nodes) {
  const int i = blockIdx.x * TPB + threadIdx.x;
  if (i < out_n4) {
    v4f z = {0.0f, 0.0f, 0.0f, 0.0f};
    ((v4f*)sums)[i] = z;
  }
  if (i < n_nodes) {
    cnt[i] = 0.0f;
  }
}

__global__ void __launch_bounds__(TPB)
mean_agg_finalize(float* __restrict__ out, const float* __restrict__ cnt,
                  int out_n4) {
  const int i = blockIdx.x * TPB + threadIdx.x;
  if (i >= out_n4) return;

  const int node = i >> 6;         // 64 float4 per row; wave-uniform
  const float c = cnt[node];       // hot in L2, reused x64 per node
  const float denom = (c > 0.0f) ? c : 1.0f;  // matches reference `where`
  v4f v = ((const v4f*)out)[i];    // hot in L2 from the atomic phase
  v.x /= denom;
  v.y /= denom;
  v.z /= denom;
  v.w /= denom;
  // Written once, never re-read: stream past L2.
  __builtin_nontemporal_store(v, (v4f*)out + i);
}

extern "C" void kernel_launch(void* const* d_in, const int* in_sizes, int n_in,
                              void* d_out, int out_size, void* d_ws, size_t ws_size,
                              hipStream_t stream) {
  const float* M     = (const float*)d_in[0];
  const int*   nodes = (const int*)d_in[1];   // values < 100000, int32 payload
  float*       out   = (float*)d_out;         // doubles as the sum accumulator
  float*       cnt   = (float*)d_ws;          // N floats of scratch

  const int D = 256;
  const int num_msgs = in_sizes[1];           // 500000
  const int n_nodes  = out_size / D;          // 100000
  const int out_n4   = out_size / 4;          // 6.4M float4

  const unsigned total4 = (unsigned)num_msgs * (D / 4);  // 32M float4

  const int zero_blocks    = (out_n4 + TPB - 1) / TPB;
  const int scatter_blocks = (int)((total4 + TPB - 1) / TPB);

  mean_agg_zero<<<zero_blocks, TPB, 0, stream>>>(out, cnt, out_n4, n_nodes);
  mean_agg_scatter<<<scatter_blocks, TPB, 0, stream>>>(M, nodes, out, cnt, total4);
  mean_agg_finalize<<<zero_blocks, TPB, 0, stream>>>(out, cnt, out_n4);
}